// KGAT_86440511799625
// MI455X (gfx1250) — compile-verified
//
#include <hip/hip_runtime.h>
#include <hip/hip_bf16.h>
#include <math.h>

#define N_NODES 100000
#define N_EDGES 1000000
#define REG_LR  1e-05f

typedef float v2f __attribute__((ext_vector_type(2)));
typedef float v8f __attribute__((ext_vector_type(8)));

__device__ __forceinline__ float lrelu(float x) {
    return x >= 0.0f ? x : 0.01f * x;
}

// ---------------------------------------------------------------------------
// Edge-weighted scatter: ego[dst] += att * node[src].  One thread per
// (edge, float2-element).  D2 = D/2 pairs per row, passed as log2 so the
// edge/pair split is a shift+mask.  Consecutive threads hit consecutive
// elements of the same edge -> coalesced gathers and coalesced f32 atomics.
// ---------------------------------------------------------------------------
__global__ __launch_bounds__(256) void kgat_edge_agg(
    const float* __restrict__ node, const float* __restrict__ att,
    const int* __restrict__ src, const int* __restrict__ dst,
    float* __restrict__ ego, int nEdges, int logTpe)
{
    long long idx = (long long)blockIdx.x * blockDim.x + threadIdx.x;
    long long total = (long long)nEdges << logTpe;
    if (idx >= total) return;
    int e = (int)(idx >> logTpe);
    int p = (int)(idx & ((1 << logTpe) - 1));
    int s = src[e], d = dst[e];
    float a = att[e];
    int D2 = 1 << logTpe;
    const float2* n2 = (const float2*)node;
    float2 v = n2[(size_t)s * D2 + p];
    float* egod = ego + ((size_t)d * D2 + p) * 2;
    atomicAdd(egod + 0, a * v.x);
    atomicAdd(egod + 1, a * v.y);
}

// ---------------------------------------------------------------------------
// Layer 0 BiInteraction GEMM via V_WMMA_F32_16X16X4_F32.
// out[n,0:32] = leaky((node+ego) @ w1^T + b1) + leaky((node*ego) @ w2^T + b2)
// One wave per 16-row block; 2 column tiles (32 outputs) and both the sum-
// and product-path accumulators live in registers (4x v8f).
// K=64 -> 16 WMMA steps of K=4.
// ---------------------------------------------------------------------------
__global__ __launch_bounds__(256) void kgat_gemm_layer0(
    const float* __restrict__ node,  // [N,64]
    const float* __restrict__ ego,   // [N,64]
    const float* __restrict__ w1,    // [32,64]
    const float* __restrict__ b1,    // [32]
    const float* __restrict__ w2,    // [32,64]
    const float* __restrict__ b2,    // [32]
    float* __restrict__ out,         // [N,32]
    int nRB)
{
    int wave = threadIdx.x >> 5;
    int lane = threadIdx.x & 31;
    int rb = blockIdx.x * 8 + wave;
    if (rb >= nRB) return;            // wave-uniform: EXEC stays all-ones

    int m  = lane & 15;               // A row within tile / B column
    int hi = lane >> 4;               // K sub-position selector
    const float* nrow = node + (size_t)(rb * 16 + m) * 64;
    const float* erow = ego  + (size_t)(rb * 16 + m) * 64;

    v8f cs0 = {}, cs1 = {}, cb0 = {}, cb1 = {};
    #pragma unroll
    for (int kk = 0; kk < 16; ++kk) {
        int kb = kk * 4 + hi * 2;
        float n0 = nrow[kb], n1 = nrow[kb + 1];
        float e0 = erow[kb], e1 = erow[kb + 1];
        v2f as = { n0 + e0, n1 + e1 };   // sum path A fragment
        v2f ab = { n0 * e0, n1 * e1 };   // product path A fragment
        v2f bs0 = { w1[m * 64 + kb],        w1[m * 64 + kb + 1] };
        v2f bs1 = { w1[(m + 16) * 64 + kb], w1[(m + 16) * 64 + kb + 1] };
        v2f bb0 = { w2[m * 64 + kb],        w2[m * 64 + kb + 1] };
        v2f bb1 = { w2[(m + 16) * 64 + kb], w2[(m + 16) * 64 + kb + 1] };
        cs0 = __builtin_amdgcn_wmma_f32_16x16x4_f32(false, as, false, bs0, (short)0, cs0, false, false);
        cs1 = __builtin_amdgcn_wmma_f32_16x16x4_f32(false, as, false, bs1, (short)0, cs1, false, false);
        cb0 = __builtin_amdgcn_wmma_f32_16x16x4_f32(false, ab, false, bb0, (short)0, cb0, false, false);
        cb1 = __builtin_amdgcn_wmma_f32_16x16x4_f32(false, ab, false, bb1, (short)0, cb1, false, false);
    }

    float bs1a = b1[m], bs1b = b1[m + 16];
    float bs2a = b2[m], bs2b = b2[m + 16];
    #pragma unroll
    for (int r = 0; r < 8; ++r) {
        int row = rb * 16 + r + hi * 8;   // C/D layout: lanes 16-31 hold M=r+8
        float o0 = lrelu(cs0[r] + bs1a) + lrelu(cb0[r] + bs2a);
        float o1 = lrelu(cs1[r] + bs1b) + lrelu(cb1[r] + bs2b);
        out[(size_t)row * 32 + m]      = o0;
        out[(size_t)row * 32 + m + 16] = o1;
    }
}

// ---------------------------------------------------------------------------
// Layer 1: [N,32] -> [N,16].  K=32 -> 8 WMMA steps; one 16-col tile.
// ---------------------------------------------------------------------------
__global__ __launch_bounds__(256) void kgat_gemm_layer1(
    const float* __restrict__ node,  // [N,32] (raw layer-0 output e1)
    const float* __restrict__ ego,   // [N,32]
    const float* __restrict__ w1,    // [16,32]
    const float* __restrict__ b1,    // [16]
    const float* __restrict__ w2,    // [16,32]
    const float* __restrict__ b2,    // [16]
    float* __restrict__ out,         // [N,16]
    int nRB)
{
    int wave = threadIdx.x >> 5;
    int lane = threadIdx.x & 31;
    int rb = blockIdx.x * 8 + wave;
    if (rb >= nRB) return;

    int m  = lane & 15;
    int hi = lane >> 4;
    const float* nrow = node + (size_t)(rb * 16 + m) * 32;
    const float* erow = ego  + (size_t)(rb * 16 + m) * 32;

    v8f cs = {}, cb = {};
    #pragma unroll
    for (int kk = 0; kk < 8; ++kk) {
        int kb = kk * 4 + hi * 2;
        float n0 = nrow[kb], n1 = nrow[kb + 1];
        float e0 = erow[kb], e1 = erow[kb + 1];
        v2f as = { n0 + e0, n1 + e1 };
        v2f ab = { n0 * e0, n1 * e1 };
        v2f bsv = { w1[m * 32 + kb], w1[m * 32 + kb + 1] };
        v2f bbv = { w2[m * 32 + kb], w2[m * 32 + kb + 1] };
        cs = __builtin_amdgcn_wmma_f32_16x16x4_f32(false, as, false, bsv, (short)0, cs, false, false);
        cb = __builtin_amdgcn_wmma_f32_16x16x4_f32(false, ab, false, bbv, (short)0, cb, false, false);
    }

    float bb1 = b1[m], bb2 = b2[m];
    #pragma unroll
    for (int r = 0; r < 8; ++r) {
        int row = rb * 16 + r + hi * 8;
        out[(size_t)row * 16 + m] = lrelu(cs[r] + bb1) + lrelu(cb[r] + bb2);
    }
}

// ---------------------------------------------------------------------------
// Row-wise L2 normalize (x / max(||x||, 1e-12)).  32/D rows per wave32;
// xor-shuffle reduction stays within each D-lane group.
// ---------------------------------------------------------------------------
template <int D>
__global__ __launch_bounds__(256) void kgat_l2norm(
    const float* __restrict__ in, float* __restrict__ out, int nRows)
{
    const int rpw = 32 / D;
    int warp = blockIdx.x * (blockDim.x >> 5) + (threadIdx.x >> 5);
    int lane = threadIdx.x & 31;
    int row = warp * rpw + lane / D;
    int col = lane % D;
    if (row >= nRows) return;
    float x = in[(size_t)row * D + col];
    float s = x * x;
    #pragma unroll
    for (int msk = D >> 1; msk; msk >>= 1) s += __shfl_xor(s, msk, 32);
    float nrm = sqrtf(s);
    out[(size_t)row * D + col] = x / fmaxf(nrm, 1e-12f);
}

// ---------------------------------------------------------------------------
// BPR scoring over 8192 triples on the virtual concat [embed64 | e1n32 | e2n16].
// out[0] += softplus(-(sp - sn)),  out[1] += REG_LR * 0.5*(|u|^2+|p|^2+|q|^2)
// ---------------------------------------------------------------------------
__global__ __launch_bounds__(256) void kgat_score(
    const float* __restrict__ embed, const float* __restrict__ e1n,
    const float* __restrict__ e2n,
    const int* __restrict__ uid, const int* __restrict__ pid,
    const int* __restrict__ nid,
    float* __restrict__ out, int nB)
{
    int t = blockIdx.x * blockDim.x + threadIdx.x;
    if (t >= nB) return;
    int u = uid[t], p = pid[t], q = nid[t];
    float sp = 0.f, sn = 0.f, ru = 0.f, rp = 0.f, rq = 0.f;

    const float* ue = embed + (size_t)u * 64;
    const float* pe = embed + (size_t)p * 64;
    const float* qe = embed + (size_t)q * 64;
    for (int d = 0; d < 64; ++d) {
        float a = ue[d], b = pe[d], c = qe[d];
        sp += a * b; sn += a * c; ru += a * a; rp += b * b; rq += c * c;
    }
    const float* u1 = e1n + (size_t)u * 32;
    const float* p1 = e1n + (size_t)p * 32;
    const float* q1 = e1n + (size_t)q * 32;
    for (int d = 0; d < 32; ++d) {
        float a = u1[d], b = p1[d], c = q1[d];
        sp += a * b; sn += a * c; ru += a * a; rp += b * b; rq += c * c;
    }
    const float* u2 = e2n + (size_t)u * 16;
    const float* p2 = e2n + (size_t)p * 16;
    const float* q2 = e2n + (size_t)q * 16;
    for (int d = 0; d < 16; ++d) {
        float a = u2[d], b = p2[d], c = q2[d];
        sp += a * b; sn += a * c; ru += a * a; rp += b * b; rq += c * c;
    }
    float z = sn - sp;                       // = -(sp - sn)
    float base = (z > 20.f) ? z : log1pf(expf(z));
    float reg = REG_LR * 0.5f * (ru + rp + rq);
    atomicAdd(&out[0], base);
    atomicAdd(&out[1], reg);
}

// ---------------------------------------------------------------------------
extern "C" void kernel_launch(void* const* d_in, const int* in_sizes, int n_in,
                              void* d_out, int out_size, void* d_ws, size_t ws_size,
                              hipStream_t stream) {
    (void)in_sizes; (void)n_in; (void)ws_size;
    const float* embed = (const float*)d_in[0];
    const float* w1_0  = (const float*)d_in[1];
    const float* b1_0  = (const float*)d_in[2];
    const float* w2_0  = (const float*)d_in[3];
    const float* b2_0  = (const float*)d_in[4];
    const float* w1_1  = (const float*)d_in[5];
    const float* b1_1  = (const float*)d_in[6];
    const float* w2_1  = (const float*)d_in[7];
    const float* b2_1  = (const float*)d_in[8];
    const float* att   = (const float*)d_in[9];
    const int*   src   = (const int*)d_in[10];
    const int*   dst   = (const int*)d_in[11];
    const int*   uids  = (const int*)d_in[12];
    const int*   pids  = (const int*)d_in[13];
    const int*   nids  = (const int*)d_in[14];
    float* out = (float*)d_out;

    // Workspace layout (floats)
    float* ws   = (float*)d_ws;
    float* ego0 = ws;                           // [N,64]
    float* e1   = ego0 + (size_t)N_NODES * 64;  // [N,32]
    float* e1n  = e1   + (size_t)N_NODES * 32;  // [N,32]
    float* ego1 = e1n  + (size_t)N_NODES * 32;  // [N,32]
    float* e2   = ego1 + (size_t)N_NODES * 32;  // [N,16]
    float* e2n  = e2   + (size_t)N_NODES * 16;  // [N,16]

    hipMemsetAsync(ego0, 0, (size_t)N_NODES * 64 * sizeof(float), stream);
    hipMemsetAsync(ego1, 0, (size_t)N_NODES * 32 * sizeof(float), stream);
    hipMemsetAsync(out, 0, (size_t)out_size * sizeof(float), stream);

    const int nRB = N_NODES / 16;               // 6250, exact

    // Layer 0 aggregation: E * 32 float2-threads
    {
        long long total = (long long)N_EDGES << 5;
        int blocks = (int)((total + 255) / 256);
        kgat_edge_agg<<<blocks, 256, 0, stream>>>(embed, att, src, dst, ego0, N_EDGES, 5);
    }
    kgat_gemm_layer0<<<(nRB + 7) / 8, 256, 0, stream>>>(
        embed, ego0, w1_0, b1_0, w2_0, b2_0, e1, nRB);
    kgat_l2norm<32><<<(N_NODES + 7) / 8, 256, 0, stream>>>(e1, e1n, N_NODES);

    // Layer 1 aggregation on raw e1: E * 16 float2-threads
    {
        long long total = (long long)N_EDGES << 4;
        int blocks = (int)((total + 255) / 256);
        kgat_edge_agg<<<blocks, 256, 0, stream>>>(e1, att, src, dst, ego1, N_EDGES, 4);
    }
    kgat_gemm_layer1<<<(nRB + 7) / 8, 256, 0, stream>>>(
        e1, ego1, w1_1, b1_1, w2_1, b2_1, e2, nRB);
    kgat_l2norm<16><<<(N_NODES / 2 + 7) / 8, 256, 0, stream>>>(e2, e2n, N_NODES);

    kgat_score<<<(8192 + 255) / 256, 256, 0, stream>>>(
        embed, e1n, e2n, uids, pids, nids, out, 8192);
}